// GCN_83099027243170
// MI455X (gfx1250) — compile-verified
//
#include <hip/hip_runtime.h>

#define NFEAT  512
#define NHID   128
#define NCLASS 40
#define NPAD   48   // NCLASS padded to 3 x 16 WMMA tiles

typedef __attribute__((ext_vector_type(2))) float v2f;
typedef __attribute__((ext_vector_type(8))) float v8f;

// ---------------------------------------------------------------------------
// GEMM1: xw[n x NHID] = x[n x NFEAT] @ w1[NFEAT x NHID]  (f32 WMMA 16x16x4)
// Block = 256 threads = 8 waves; wave w owns C tile (blockIdx.x*16, w*16).
// A layout (16x4 f32): lanes 0-15 -> M=lane, K offsets {0,1};
//                      lanes 16-31 -> M=lane-16, K offsets {2,3}.
// B layout (4x16 f32): lanes 0-15 -> N=lane, K offsets {0,1};
//                      lanes 16-31 -> N=lane-16, K offsets {2,3}.
// C layout: VGPR v -> row (v + 8*(lane>=16)), col lane%16.
// ---------------------------------------------------------------------------
__global__ void gcn_gemm1(const float* __restrict__ x,
                          const float* __restrict__ w1,
                          float* __restrict__ xw, int nrows) {
  const int lane = threadIdx.x & 31;
  const int wave = threadIdx.x >> 5;     // 0..7 -> N tile
  const int half = lane >> 4;            // 0 or 1
  const int lm   = lane & 15;
  const int m0   = blockIdx.x * 16;
  const int n0   = wave * 16;

  int mrow = m0 + lm;
  if (mrow >= nrows) mrow = nrows - 1;   // clamp (tail tile); EXEC stays all-1s

  const float* __restrict__ xr = x  + (size_t)mrow * NFEAT + half * 2;
  const float* __restrict__ wc = w1 + (size_t)(half * 2) * NHID + n0 + lm;

  v8f c = {};
  for (int k = 0; k < NFEAT; k += 4) {
    __builtin_prefetch(xr + k + 32, 0, 3);   // global_prefetch_b8, WGP scope
    v2f a, b;
    a[0] = xr[k];
    a[1] = xr[k + 1];
    b[0] = wc[(size_t)k * NHID];
    b[1] = wc[(size_t)(k + 1) * NHID];
    c = __builtin_amdgcn_wmma_f32_16x16x4_f32(false, a, false, b,
                                              (short)0, c, false, false);
  }

  float* __restrict__ op = xw + (size_t)(m0 + half * 8) * NHID + n0 + lm;
  if (m0 + 16 <= nrows) {
    // full tile: unguarded coalesced stores
#pragma unroll
    for (int v = 0; v < 8; ++v)
      op[(size_t)v * NHID] = c[v];
  } else {
#pragma unroll
    for (int v = 0; v < 8; ++v)
      if (m0 + v + half * 8 < nrows) op[(size_t)v * NHID] = c[v];
  }
}

// ---------------------------------------------------------------------------
// GEMM2: hw[n x NCLASS] = h[n x NHID] @ w2p[NHID x NPAD] (f32 WMMA 16x16x4)
// Block = 96 threads = 3 waves (3 N tiles of 16, cols 40..47 are zero pad).
// ---------------------------------------------------------------------------
__global__ void gcn_gemm2(const float* __restrict__ h,
                          const float* __restrict__ w2p,
                          float* __restrict__ hw, int nrows) {
  const int lane = threadIdx.x & 31;
  const int wave = threadIdx.x >> 5;     // 0..2 -> N tile
  const int half = lane >> 4;
  const int lm   = lane & 15;
  const int m0   = blockIdx.x * 16;
  const int n0   = wave * 16;

  int mrow = m0 + lm;
  if (mrow >= nrows) mrow = nrows - 1;

  const float* __restrict__ hr = h   + (size_t)mrow * NHID + half * 2;
  const float* __restrict__ wc = w2p + (size_t)(half * 2) * NPAD + n0 + lm;

  v8f c = {};
  for (int k = 0; k < NHID; k += 4) {
    v2f a, b;
    a[0] = hr[k];
    a[1] = hr[k + 1];
    b[0] = wc[(size_t)k * NPAD];
    b[1] = wc[(size_t)(k + 1) * NPAD];
    c = __builtin_amdgcn_wmma_f32_16x16x4_f32(false, a, false, b,
                                              (short)0, c, false, false);
  }

  const int col = n0 + lm;
  float* __restrict__ op = hw + (size_t)(m0 + half * 8) * NCLASS + col;
  if (col < NCLASS) {                       // lane-uniform across the 8 stores
    if (m0 + 16 <= nrows) {
#pragma unroll
      for (int v = 0; v < 8; ++v)
        op[(size_t)v * NCLASS] = c[v];
    } else {
#pragma unroll
      for (int v = 0; v < 8; ++v)
        if (m0 + v + half * 8 < nrows) op[(size_t)v * NCLASS] = c[v];
    }
  }
}

// Zero-pad w2 (NHID x NCLASS) into w2p (NHID x NPAD)
__global__ void gcn_pad_w2(const float* __restrict__ w2, float* __restrict__ w2p) {
  int idx = blockIdx.x * blockDim.x + threadIdx.x;
  if (idx >= NHID * NPAD) return;
  int r = idx / NPAD, cc = idx - r * NPAD;
  w2p[idx] = (cc < NCLASS) ? w2[r * NCLASS + cc] : 0.0f;
}

// SpMM layer 1: dst[row[e], t] += w[e] * src[col[e], t], t in [0,NHID)
__global__ void gcn_spmm_hid(const int* __restrict__ er, const int* __restrict__ ec,
                             const float* __restrict__ ew,
                             const float* __restrict__ src, float* __restrict__ dst) {
  const int e = blockIdx.x;
  const int t = threadIdx.x;                 // 0..NHID-1
  const int r = er[e], cl = ec[e];
  const float v = ew[e] * src[(size_t)cl * NHID + t];
  atomicAdd(dst + (size_t)r * NHID + t, v);  // global_atomic_add_f32
}

// SpMM layer 2: out[row[e], t] += w[e] * hw[col[e], t], t in [0,NCLASS)
__global__ void gcn_spmm_out(const int* __restrict__ er, const int* __restrict__ ec,
                             const float* __restrict__ ew,
                             const float* __restrict__ hw, float* __restrict__ out) {
  const int e = blockIdx.x;
  const int t = threadIdx.x;                 // block = 64, active t < NCLASS
  if (t < NCLASS) {
    const int r = er[e], cl = ec[e];
    const float v = ew[e] * hw[(size_t)cl * NCLASS + t];
    atomicAdd(out + (size_t)r * NCLASS + t, v);
  }
}

// h = relu(h + b1)
__global__ void gcn_bias_relu(float* __restrict__ h, const float* __restrict__ b1,
                              size_t total) {
  size_t i = (size_t)blockIdx.x * blockDim.x + threadIdx.x;
  if (i < total) {
    float v = h[i] + b1[i & (NHID - 1)];
    h[i] = v > 0.0f ? v : 0.0f;
  }
}

// out[i] = b2[i % NCLASS]  (bias pre-init; spmm2 atomics accumulate on top)
__global__ void gcn_init_out(float* __restrict__ out, const float* __restrict__ b2,
                             size_t total) {
  size_t i = (size_t)blockIdx.x * blockDim.x + threadIdx.x;
  if (i < total) out[i] = b2[i % NCLASS];
}

extern "C" void kernel_launch(void* const* d_in, const int* in_sizes, int n_in,
                              void* d_out, int out_size, void* d_ws, size_t ws_size,
                              hipStream_t stream) {
  const float* x  = (const float*)d_in[0];
  const int*   er = (const int*)  d_in[1];
  const int*   ec = (const int*)  d_in[2];
  const float* ew = (const float*)d_in[3];
  const float* w1 = (const float*)d_in[4];
  const float* b1 = (const float*)d_in[5];
  const float* w2 = (const float*)d_in[6];
  const float* b2 = (const float*)d_in[7];
  float* out = (float*)d_out;

  const int n  = in_sizes[0] / NFEAT;   // 50000
  const int ne = in_sizes[1];           // 800000

  // Workspace layout (floats): xw | h | hw | w2p  (~60 MB total)
  float* xw  = (float*)d_ws;
  float* h   = xw + (size_t)n * NHID;
  float* hw  = h  + (size_t)n * NHID;
  float* w2p = hw + (size_t)n * NCLASS;

  const int mtiles = (n + 15) / 16;

  // h is accumulated by atomics -> must be zeroed every call.
  hipMemsetAsync(h, 0, (size_t)n * NHID * sizeof(float), stream);

  gcn_gemm1<<<mtiles, 256, 0, stream>>>(x, w1, xw, n);
  gcn_pad_w2<<<(NHID * NPAD + 255) / 256, 256, 0, stream>>>(w2, w2p);
  gcn_spmm_hid<<<ne, NHID, 0, stream>>>(er, ec, ew, xw, h);

  {
    size_t tot = (size_t)n * NHID;
    gcn_bias_relu<<<(unsigned)((tot + 255) / 256), 256, 0, stream>>>(h, b1, tot);
  }

  gcn_gemm2<<<mtiles, 96, 0, stream>>>(h, w2p, hw, n);

  {
    size_t tot = (size_t)n * NCLASS;
    gcn_init_out<<<(unsigned)((tot + 255) / 256), 256, 0, stream>>>(out, b2, tot);
  }

  gcn_spmm_out<<<ne, 64, 0, stream>>>(er, ec, ew, hw, out);
}